// Conv1dMapper_28870770164404
// MI455X (gfx1250) — compile-verified
//
#include <hip/hip_runtime.h>
#include <hip/hip_bf16.h>

typedef __attribute__((ext_vector_type(2))) float v2f;
typedef __attribute__((ext_vector_type(8))) float v8f;

#define EPSV 1e-5f

// ---------------------------------------------------------------------------
// Conv1d (VALID, kernel 3) + bias + ReLU, layout (N, C, L)
// ---------------------------------------------------------------------------
__global__ __launch_bounds__(256) void conv_relu_kernel(
    const float* __restrict__ x, const float* __restrict__ w,
    const float* __restrict__ bias, float* __restrict__ y,
    int Cin, int Lin, int Cout, int Lout, int total) {
  int i = blockIdx.x * 256 + threadIdx.x;
  if (i >= total) return;
  int p = i % Lout;
  int t = i / Lout;
  int co = t % Cout;
  int n = t / Cout;
  float acc = bias[co];
  const float* xr = x + (size_t)n * Cin * Lin;
  const float* wr = w + (size_t)co * Cin * 3;
  for (int ci = 0; ci < Cin; ++ci) {
    const float* xp = xr + ci * Lin + p;
    const float* wp = wr + ci * 3;
    acc += xp[0] * wp[0] + xp[1] * wp[1] + xp[2] * wp[2];
  }
  y[i] = fmaxf(acc, 0.f);
}

// ---------------------------------------------------------------------------
// Per-channel batchnorm over (batch=64, length): one block per channel.
// ---------------------------------------------------------------------------
__global__ __launch_bounds__(256) void bn_channel_kernel(
    float* __restrict__ y, const float* __restrict__ g,
    const float* __restrict__ b, int C, int Lp) {
  const int c = blockIdx.x;
  const int Np = 64 * Lp;
  __shared__ float red[256];
  __shared__ float red2[256];
  float s = 0.f, sq = 0.f;
  for (int i = threadIdx.x; i < Np; i += 256) {
    int n = i / Lp, p = i % Lp;
    float v = y[((size_t)n * C + c) * Lp + p];
    s += v; sq += v * v;
  }
  red[threadIdx.x] = s;
  red2[threadIdx.x] = sq;
  __syncthreads();
  for (int st = 128; st > 0; st >>= 1) {
    if (threadIdx.x < st) {
      red[threadIdx.x] += red[threadIdx.x + st];
      red2[threadIdx.x] += red2[threadIdx.x + st];
    }
    __syncthreads();
  }
  float mean = red[0] / (float)Np;
  float var = red2[0] / (float)Np - mean * mean;
  float rstd = rsqrtf(var + EPSV);
  float gg = g[c] * rstd;
  float bb = b[c];
  for (int i = threadIdx.x; i < Np; i += 256) {
    int n = i / Lp, p = i % Lp;
    size_t idx = ((size_t)n * C + c) * Lp + p;
    y[idx] = (y[idx] - mean) * gg + bb;
  }
}

// ---------------------------------------------------------------------------
// FC GEMM via V_WMMA_F32_16X16X4_F32, M=64 (whole batch in one workgroup),
// N tiled by 32 per workgroup, K streamed in 128-wide LDS chunks with
// register double-buffering and dual accumulators (breaks the WMMA D->C
// dependence chain). Optionally fuses bias+ReLU+bn_f (batch mean/var).
//   out[m, n] = act[m, :] . wgt[n, :]   (wgt row-major (N, K))
// Wave layout: 8 waves; wave w -> M-tile (w&3)*16, N-subtile (w>>2)*16.
// ---------------------------------------------------------------------------
template <bool FUSE_BN>
__global__ __launch_bounds__(256) void fc_wmma_kernel(
    const float* __restrict__ act, const float* __restrict__ wgt,
    const float* __restrict__ bias, const float* __restrict__ g,
    const float* __restrict__ bb, float* __restrict__ out, int K, int N) {
  constexpr int KC = 128;  // K chunk
  constexpr int AS = 132;  // LDS row stride (floats): 16B aligned, bank-spread
  constexpr int BS = 132;
  __shared__ float sA[64 * AS];   // 64 rows of activations x KC
  __shared__ float sB[32 * BS];   // 32 rows of weights      x KC

  const int t = threadIdx.x;
  const int lane = t & 31;
  const int lo = lane & 15;
  const int hi = lane >> 4;
  const int wv = t >> 5;
  const int mi = wv & 3;   // M tile 0..3 (rows mi*16 .. mi*16+15)
  const int ni = wv >> 2;  // N subtile 0..1
  const int n0 = blockIdx.x * 32;

  const int nch = (K + KC - 1) / KC;

  float4 ra[8];
  float4 rb[4];

  auto load_chunk = [&](int ch) {
    const int kbase = ch * KC;
    const bool fullk = (kbase + KC) <= K;
#pragma unroll
    for (int i = 0; i < 8; ++i) {
      int f = t + i * 256;         // float4 id, 0..2047
      int row = f >> 5;            // activation row 0..63 (32 float4 per row)
      int c4 = (f & 31) << 2;      // k offset within chunk
      int k = kbase + c4;
      float4 v = make_float4(0.f, 0.f, 0.f, 0.f);
      const float* p = act + (size_t)row * K;
      if (fullk) {
        v = *(const float4*)(p + k);
      } else {
        if (k + 0 < K) v.x = p[k + 0];
        if (k + 1 < K) v.y = p[k + 1];
        if (k + 2 < K) v.z = p[k + 2];
        if (k + 3 < K) v.w = p[k + 3];
      }
      ra[i] = v;
    }
#pragma unroll
    for (int i = 0; i < 4; ++i) {
      int f = t + i * 256;         // float4 id, 0..1023
      int r = f >> 5;              // weight row 0..31
      int c4 = (f & 31) << 2;
      int n = n0 + r;
      int k = kbase + c4;
      float4 v = make_float4(0.f, 0.f, 0.f, 0.f);
      if (n < N) {
        const float* p = wgt + (size_t)n * K;
        if (fullk) {
          v = *(const float4*)(p + k);
        } else {
          if (k + 0 < K) v.x = p[k + 0];
          if (k + 1 < K) v.y = p[k + 1];
          if (k + 2 < K) v.z = p[k + 2];
          if (k + 3 < K) v.w = p[k + 3];
        }
      }
      rb[i] = v;
    }
  };

  v8f acc0 = {0.f, 0.f, 0.f, 0.f, 0.f, 0.f, 0.f, 0.f};
  v8f acc1 = {0.f, 0.f, 0.f, 0.f, 0.f, 0.f, 0.f, 0.f};

  load_chunk(0);
  for (int ch = 0; ch < nch; ++ch) {
    __syncthreads();   // previous chunk's LDS reads done
#pragma unroll
    for (int i = 0; i < 8; ++i) {
      int f = t + i * 256;
      int row = f >> 5;
      int c4 = (f & 31) << 2;
      *(float4*)&sA[row * AS + c4] = ra[i];
    }
#pragma unroll
    for (int i = 0; i < 4; ++i) {
      int f = t + i * 256;
      int r = f >> 5;
      int c4 = (f & 31) << 2;
      *(float4*)&sB[r * BS + c4] = rb[i];
    }
    __syncthreads();   // LDS visible
    if (ch + 1 < nch) load_chunk(ch + 1);  // prefetch overlaps WMMA below

    // A frag: lane (lo,hi) -> row mi*16+lo, K pair {2*hi, 2*hi+1} (+4 per kk)
    // B frag: lane (lo,hi) -> wgt row ni*16+lo (column n), same K pair
    const float* pA = &sA[(mi * 16 + lo) * AS + 2 * hi];
    const float* pB = &sB[(ni * 16 + lo) * BS + 2 * hi];
#pragma unroll
    for (int kk = 0; kk < 32; kk += 2) {
      v2f a0 = *(const v2f*)(pA + kk * 4);
      v2f b0 = *(const v2f*)(pB + kk * 4);
      v2f a1 = *(const v2f*)(pA + kk * 4 + 4);
      v2f b1 = *(const v2f*)(pB + kk * 4 + 4);
      acc0 = __builtin_amdgcn_wmma_f32_16x16x4_f32(
          false, a0, false, b0, (short)0, acc0, false, false);
      acc1 = __builtin_amdgcn_wmma_f32_16x16x4_f32(
          false, a1, false, b1, (short)0, acc1, false, false);
    }
  }

  v8f acc;
#pragma unroll
  for (int v = 0; v < 8; ++v) acc[v] = acc0[v] + acc1[v];

  // D layout: lane lo = local col, element v -> local row v + 8*hi
  const int col_l = ni * 16 + lo;
  const int col = n0 + col_l;
  const float biasv = (col < N) ? bias[col] : 0.f;

  if (FUSE_BN) {
#pragma unroll
    for (int v = 0; v < 8; ++v) acc[v] = fmaxf(acc[v] + biasv, 0.f);
    __syncthreads();                 // staging LDS now dead, reuse
    constexpr int TS = 33;
#pragma unroll
    for (int v = 0; v < 8; ++v) {
      int row = mi * 16 + v + 8 * hi;
      sA[row * TS + col_l] = acc[v];
    }
    __syncthreads();
    if (t < 32) {                    // per-column stats over the 64-row batch
      float s = 0.f, sq = 0.f;
#pragma unroll 8
      for (int r = 0; r < 64; ++r) {
        float xv = sA[r * TS + t];
        s += xv; sq += xv * xv;
      }
      float mean = s * (1.f / 64.f);
      float var = sq * (1.f / 64.f) - mean * mean;
      sB[t] = mean;
      sB[32 + t] = rsqrtf(var + EPSV);
    }
    __syncthreads();
    if (col < N) {
      float mean = sB[col_l];
      float gv = g[col] * sB[32 + col_l];
      float bv2 = bb[col];
#pragma unroll
      for (int v = 0; v < 8; ++v) {
        int row = mi * 16 + v + 8 * hi;
        out[(size_t)row * N + col] = (acc[v] - mean) * gv + bv2;
      }
    }
  } else {
    if (col < N) {
#pragma unroll
      for (int v = 0; v < 8; ++v) {
        int row = mi * 16 + v + 8 * hi;
        out[(size_t)row * N + col] = acc[v] + biasv;
      }
    }
  }
}

// ---------------------------------------------------------------------------
// Output mangling: zero channels, ones channel, piecewise-linear interp.
// h: (64, 1200) fc3 output; out: (64, 12, 100)
// ---------------------------------------------------------------------------
__global__ __launch_bounds__(256) void finalize_kernel(
    const float* __restrict__ h, float* __restrict__ out) {
  int i = blockIdx.x * 256 + threadIdx.x;
  if (i >= 64 * 12 * 100) return;
  int p = i % 100;
  int t = i / 100;
  int c = t % 12;
  int n = t / 12;
  const unsigned zmask = (1u << 1) | (1u << 4) | (1u << 6) | (1u << 7) | (1u << 9);
  float val;
  if (zmask & (1u << c)) {
    val = 0.f;
  } else if (c == 5) {
    val = 1.f;
  } else {
    const float* hr = h + (size_t)n * 1200 + c * 100;
    if (p == 0 || p == 1 || p == 33 || p == 66 || p == 99) {
      val = hr[p];
    } else {
      int j = p / 33;                       // 0,1,2
      int l = (j == 0) ? 1 : ((j == 1) ? 33 : 66);
      int r = (j == 0) ? 33 : ((j == 1) ? 66 : 99);
      float a = (float)(p - l) * (1.f / 33.f);
      val = a * hr[l] + (1.f - a) * hr[r];
    }
  }
  out[i] = val;
}

// ---------------------------------------------------------------------------
extern "C" void kernel_launch(void* const* d_in, const int* in_sizes, int n_in,
                              void* d_out, int out_size, void* d_ws, size_t ws_size,
                              hipStream_t stream) {
  const float* x    = (const float*)d_in[0];
  const float* c1w  = (const float*)d_in[1];
  const float* c1b  = (const float*)d_in[2];
  const float* g1   = (const float*)d_in[3];
  const float* b1   = (const float*)d_in[4];
  const float* c2w  = (const float*)d_in[5];
  const float* c2b  = (const float*)d_in[6];
  const float* g2   = (const float*)d_in[7];
  const float* b2   = (const float*)d_in[8];
  const float* c3w  = (const float*)d_in[9];
  const float* c3b  = (const float*)d_in[10];
  const float* g3   = (const float*)d_in[11];
  const float* b3   = (const float*)d_in[12];
  const float* fc1w = (const float*)d_in[13];
  const float* fc1b = (const float*)d_in[14];
  const float* g4   = (const float*)d_in[15];
  const float* b4   = (const float*)d_in[16];
  const float* fc2w = (const float*)d_in[17];
  const float* fc2b = (const float*)d_in[18];
  const float* g5   = (const float*)d_in[19];
  const float* b5   = (const float*)d_in[20];
  const float* fc3w = (const float*)d_in[21];
  const float* fc3b = (const float*)d_in[22];
  float* out = (float*)d_out;
  float* ws  = (float*)d_ws;

  // Workspace layout (floats). a1 overlays h1/h2 (dead once h3 exists).
  float* h1 = ws;                  // 64*64*98  = 401408
  float* h2 = h1 + 64 * 64 * 98;   // 64*64*96  = 393216
  float* h3 = h2 + 64 * 64 * 96;   // 64*64*94  = 385024  (flat = (64, 6016))
  float* a1 = ws;                  // 64*10000  = 640000  (over h1+h2 = 794624)
  float* a2 = h3 + 64 * 64 * 94;   // 64*10000  = 640000
  float* hf = a2 + 64 * 10000;     // 64*1200   = 76800
  (void)ws_size; (void)in_sizes; (void)n_in; (void)out_size;

  // conv1 -> bn1
  {
    int total = 64 * 64 * 98;
    conv_relu_kernel<<<(total + 255) / 256, 256, 0, stream>>>(
        x, c1w, c1b, h1, 3, 100, 64, 98, total);
    bn_channel_kernel<<<64, 256, 0, stream>>>(h1, g1, b1, 64, 98);
  }
  // conv2 -> bn2
  {
    int total = 64 * 64 * 96;
    conv_relu_kernel<<<(total + 255) / 256, 256, 0, stream>>>(
        h1, c2w, c2b, h2, 64, 98, 64, 96, total);
    bn_channel_kernel<<<64, 256, 0, stream>>>(h2, g2, b2, 64, 96);
  }
  // conv3 -> bn3
  {
    int total = 64 * 64 * 94;
    conv_relu_kernel<<<(total + 255) / 256, 256, 0, stream>>>(
        h2, c3w, c3b, h3, 64, 96, 64, 94, total);
    bn_channel_kernel<<<64, 256, 0, stream>>>(h3, g3, b3, 64, 94);
  }

  // fc1 + relu + bn4 (fused)
  fc_wmma_kernel<true><<<(10000 + 31) / 32, 256, 0, stream>>>(
      h3, fc1w, fc1b, g4, b4, a1, 6016, 10000);
  // fc2 + relu + bn5 (fused)
  fc_wmma_kernel<true><<<(10000 + 31) / 32, 256, 0, stream>>>(
      a1, fc2w, fc2b, g5, b5, a2, 10000, 10000);
  // fc3 (bias only)
  fc_wmma_kernel<false><<<(1200 + 31) / 32, 256, 0, stream>>>(
      a2, fc3w, fc3b, fc3b, fc3b, hf, 10000, 1200);

  // channel rules + interpolation
  finalize_kernel<<<(64 * 12 * 100 + 255) / 256, 256, 0, stream>>>(hf, out);
}